// SpGraphAttentionLayer_58265526338328
// MI455X (gfx1250) — compile-verified
//
#include <hip/hip_runtime.h>
#include <hip/hip_bf16.h>
#include <cstdint>

#define N_NODES 50000
#define N_EDGES 800000
#define F_IN    128
#define F_OUT   64
#define ALPHA   0.2f

typedef __attribute__((ext_vector_type(2))) float v2f;
typedef __attribute__((ext_vector_type(8))) float v8f;

// -----------------------------------------------------------------------------
// Kernel 1: h = X @ W  via V_WMMA_F32_16X16X4_F32
// grid = (N_NODES/16, F_OUT/16), block = 32 (one wave per 16x16 tile)
// A 16x4 layout: lanes 0-15 hold row M=lane, {K,K+1}; lanes 16-31 hold {K+2,K+3}
// B 4x16 layout: VGPR0 = rows K (lanes 0-15) / K+2 (lanes 16-31); VGPR1 = K+1 / K+3
// C/D 16x16:     VGPR v = row (v + 8*half), col = lane%16
// -----------------------------------------------------------------------------
__global__ void gat_gemm_wmma_f32(const float* __restrict__ X,
                                  const float* __restrict__ W,
                                  float* __restrict__ H) {
  const int lane = threadIdx.x & 31;
  const int half = lane >> 4;      // 0 or 1
  const int lr   = lane & 15;
  const int m0   = blockIdx.x * 16;
  const int n0   = blockIdx.y * 16;

  const float* __restrict__ arow = X + (size_t)(m0 + lr) * F_IN + 2 * half;

  v8f acc = {};
#pragma unroll
  for (int k0 = 0; k0 < F_IN; k0 += 4) {
    const int ka = k0 + 2 * half;
    v2f a;
    a.x = arow[k0 + 0];
    a.y = arow[k0 + 1];
    v2f b;
    b.x = W[(ka + 0) * F_OUT + n0 + lr];
    b.y = W[(ka + 1) * F_OUT + n0 + lr];
    acc = __builtin_amdgcn_wmma_f32_16x16x4_f32(
        /*neg_a=*/false, a, /*neg_b=*/false, b,
        /*c_mod=*/(short)0, acc, /*reuse_a=*/false, /*reuse_b=*/false);
  }

#pragma unroll
  for (int v = 0; v < 8; ++v) {
    H[(size_t)(m0 + v + 8 * half) * F_OUT + n0 + lr] = acc[v];
  }
}

// -----------------------------------------------------------------------------
// Kernel 2: per-node attention scores: ssrc[i] = h[i,:]·a[0:64], sdst = h·a[64:128]
// One wave per node, shuffle reduction (wave32).
// -----------------------------------------------------------------------------
__global__ void gat_scores(const float* __restrict__ H,
                           const float* __restrict__ a,
                           float* __restrict__ ssrc,
                           float* __restrict__ sdst) {
  const int wave = (int)((blockIdx.x * (unsigned)blockDim.x + threadIdx.x) >> 5);
  const int lane = threadIdx.x & 31;
  if (wave >= N_NODES) return;

  const float h0 = H[(size_t)wave * F_OUT + lane];
  const float h1 = H[(size_t)wave * F_OUT + lane + 32];
  float ps = fmaf(h0, a[lane],      h1 * a[lane + 32]);
  float pd = fmaf(h0, a[64 + lane], h1 * a[96 + lane]);

#pragma unroll
  for (int off = 16; off > 0; off >>= 1) {
    ps += __shfl_down(ps, off, 32);
    pd += __shfl_down(pd, off, 32);
  }
  if (lane == 0) {
    ssrc[wave] = ps;
    sdst[wave] = pd;
  }
}

// -----------------------------------------------------------------------------
// Kernel 3: edge scatter. One wave per edge; 64 features across 32 lanes (x2).
// edge_e = exp(-leaky_relu(ssrc[s]+sdst[d])) * adj;  atomically accumulate
// rowsum[s] += edge_e;  hprime[s,:] += edge_e * h[d,:].   All L2-resident.
// -----------------------------------------------------------------------------
__global__ void gat_edges(const float* __restrict__ H,
                          const float* __restrict__ ssrc,
                          const float* __restrict__ sdst,
                          const float* __restrict__ adj,
                          const long long* __restrict__ ei,
                          float* __restrict__ rowsum,
                          float* __restrict__ hprime) {
  const long long e = (long long)((blockIdx.x * (unsigned long long)blockDim.x + threadIdx.x) >> 5);
  const int lane = threadIdx.x & 31;
  if (e >= N_EDGES) return;

  const int s = (int)ei[e];
  const int d = (int)ei[N_EDGES + e];

  const float sc   = ssrc[s] + sdst[d];
  const float lrel = sc > 0.0f ? sc : ALPHA * sc;
  const float w    = __expf(-lrel) * adj[e];

  if (lane == 0) {
    atomicAdd(&rowsum[s], w);
  }
  const float g0 = H[(size_t)d * F_OUT + lane];
  const float g1 = H[(size_t)d * F_OUT + lane + 32];
  atomicAdd(&hprime[(size_t)s * F_OUT + lane],      w * g0);
  atomicAdd(&hprime[(size_t)s * F_OUT + lane + 32], w * g1);
}

// -----------------------------------------------------------------------------
// Kernel 4: out = elu(hprime / rowsum[node])  (in place on d_out)
// -----------------------------------------------------------------------------
__global__ void gat_finalize(const float* __restrict__ rowsum,
                             float* __restrict__ out) {
  const int idx = (int)(blockIdx.x * blockDim.x + threadIdx.x);
  if (idx >= N_NODES * F_OUT) return;
  const float v = out[idx] / rowsum[idx >> 6];
  out[idx] = v > 0.0f ? v : (__expf(v) - 1.0f);
}

// -----------------------------------------------------------------------------
// Launch
// d_in: [input(50000x128 f32), W(128x64 f32), a(1x128 f32),
//        adj_vals(800000 f32), edge_index(2x800000 i64)]
// d_out: 50000x64 f32. d_out doubles as the h_prime accumulator.
// -----------------------------------------------------------------------------
extern "C" void kernel_launch(void* const* d_in, const int* in_sizes, int n_in,
                              void* d_out, int out_size, void* d_ws, size_t ws_size,
                              hipStream_t stream) {
  const float*     X   = (const float*)d_in[0];
  const float*     W   = (const float*)d_in[1];
  const float*     a   = (const float*)d_in[2];
  const float*     adj = (const float*)d_in[3];
  const long long* ei  = (const long long*)d_in[4];
  float* out = (float*)d_out;

  float* H      = (float*)d_ws;               // N_NODES * F_OUT
  float* ssrc   = H + (size_t)N_NODES * F_OUT; // N_NODES
  float* sdst   = ssrc + N_NODES;              // N_NODES
  float* rowsum = sdst + N_NODES;              // N_NODES

  hipMemsetAsync(out,    0, sizeof(float) * (size_t)N_NODES * F_OUT, stream);
  hipMemsetAsync(rowsum, 0, sizeof(float) * (size_t)N_NODES, stream);

  gat_gemm_wmma_f32<<<dim3(N_NODES / 16, F_OUT / 16), 32, 0, stream>>>(X, W, H);

  {
    const long long threads = (long long)N_NODES * 32;
    gat_scores<<<(unsigned)((threads + 255) / 256), 256, 0, stream>>>(H, a, ssrc, sdst);
  }
  {
    const long long threads = (long long)N_EDGES * 32;
    gat_edges<<<(unsigned)((threads + 255) / 256), 256, 0, stream>>>(H, ssrc, sdst, adj, ei,
                                                                    rowsum, out);
  }
  gat_finalize<<<(N_NODES * F_OUT + 255) / 256, 256, 0, stream>>>(rowsum, out);
}